// RIPShiftedLoss_5231270167086
// MI455X (gfx1250) — compile-verified
//
#include <hip/hip_runtime.h>
#include <hip/hip_bf16.h>
#include <math.h>
#include <float.h>

typedef __attribute__((ext_vector_type(2))) float v2f;
typedef __attribute__((ext_vector_type(8))) float v8f;

#define TS     32     // tile size s
#define NSH    17     // shifts per axis (-8..8)
#define SLAB   48     // slab rows/cols
#define SLABP  49     // padded LDS stride for slab
#define SATP   50     // padded LDS stride for SAT (49x49 used)
#define NANG   11     // angles -5..5
#define APADP  34     // aPad stride: 34 floats = 136B (8B multiple -> b64-able pairs)

__device__ __forceinline__ float wave_sum32(float v) {
#pragma unroll
  for (int m = 16; m >= 1; m >>= 1) v += __shfl_xor(v, m, 32);
  return v;
}
__device__ __forceinline__ float wave_min32(float v) {
#pragma unroll
  for (int m = 16; m >= 1; m >>= 1) v = fminf(v, __shfl_xor(v, m, 32));
  return v;
}

// One wave per (batch, tile). Computes sub_min[bt] = min(s_min, r_min).
__global__ __launch_bounds__(32)
void rip_tile_kernel(const float* __restrict__ fm1, const float* __restrict__ fm2,
                     float* __restrict__ subMin)
{
  // Zero-padded band buffer: logical rows -16..47 stored at 0..63.
  // Rows 16..47 hold the fm1 tile; rest stay zero -> unconditional band reads.
  __shared__ float aPad[64 * APADP];
  __shared__ float slab[SLAB * SLABP];     // fm2 48x48 slab
  __shared__ float sat[49 * SATP];         // SAT of slab^2 (49x49, zero border)

  const int bt   = blockIdx.x;
  const int b    = bt >> 4, tile = bt & 15;
  const int ty   = tile >> 2, tx = tile & 3;
  const int lane = threadIdx.x;
  const int half = lane >> 4, lid = lane & 15;

  const float* img1 = fm1 + b * (128 * 128);
  const float* img2 = fm2 + b * (128 * 128);

  const int rbase0 = min(max(ty * 32 - 8, 0), 96);   // first slab row in fm2
  const int cbase0 = min(max(tx * 32 - 8, 0), 96);   // first slab col in fm2
  const int rc0 = ty * 32 - rbase0;                  // central tile offset in slab
  const int cc0 = tx * 32 - cbase0;

  // ---- zero aPad, then load fm1 tile into rows 16..47; fold in Sa ----
  for (int i = lane; i < 64 * APADP; i += 32) aPad[i] = 0.f;
  float sa = 0.f;
  for (int it = 0; it < 32; ++it) {
    int idx = it * 32 + lane;
    int r = idx >> 5, c = idx & 31;
    float v = img1[(ty * 32 + r) * 128 + (tx * 32 + c)];
    aPad[(r + 16) * APADP + c] = v;
    sa += v * v;
  }
  sa = wave_sum32(sa);   // Sa on all lanes

  // ---- load fm2 slab 48x48 (edge-clamped; clamped cells never used) ----
  for (int it = 0; it < 72; ++it) {
    int idx = it * 32 + lane;
    int r = idx / 48, c = idx - r * 48;
    int gr = min(rbase0 + r, 127), gc = min(cbase0 + c, 127);
    slab[r * SLABP + c] = img2[gr * 128 + gc];
  }
  __syncthreads();

  // ---- SAT of slab^2: sat[r][c] = sum over slab[0..r-1][0..c-1]^2 ----
  for (int rr = lane; rr < SLAB; rr += 32) {
    float run = 0.f;
    sat[(rr + 1) * SATP + 0] = 0.f;
    for (int c = 0; c < SLAB; ++c) {
      float v = slab[rr * SLABP + c];
      run += v * v;
      sat[(rr + 1) * SATP + (c + 1)] = run;
    }
  }
  for (int c = lane; c < 49; c += 32) sat[c] = 0.f;   // top zero row
  __syncthreads();
  for (int cc = lane + 1; cc < 49; cc += 32) {        // column prefix
    float run = 0.f;
    for (int r = 1; r <= SLAB; ++r) {
      run += sat[r * SATP + cc];
      sat[r * SATP + cc] = run;
    }
  }
  __syncthreads();

  // ---- banded GEMM via V_WMMA_F32_16X16X4_F32 ----
  // C[dy,dx] = sum_{R=0..47, c=0..31} Aband[dy,(R,c)] * slab[R][coloff(dx)+c]
  // Aband[dy,(R,c)] = aPad[(R - rowoff(dy) + 16)][c]   (zeros pad the band)
  const int ro0 = min(max(ty * 32 + lid - 8, 0), 96) - rbase0;  // dy = 0..15 (per lane)
  const int ro1 = min(max(ty * 32 + 8, 0), 96) - rbase0;        // dy = 16 (wave-uniform)
  const int co0 = min(max(tx * 32 + lid - 8, 0), 96) - cbase0;  // dx = 0..15 (per lane)
  const int co1 = min(max(tx * 32 + 8, 0), 96) - cbase0;        // dx = 16 (wave-uniform)

  const int aRow0 = 16 - ro0;   // aPad row for R = 0
  const int aRow1 = 16 - ro1;
  const int h2 = half * 2;      // K sub-pair select per lane half

  v8f c00 = {}, c01 = {}, c10 = {}, c11 = {};
  for (int R = 0; R < SLAB; ++R) {
    const float* a0p = &aPad[(R + aRow0) * APADP + h2];
    const float* a1p = &aPad[(R + aRow1) * APADP + h2];
    const float* b0p = &slab[R * SLABP + co0 + h2];
    const float* b1p = &slab[R * SLABP + co1 + h2];
#pragma unroll
    for (int q = 0; q < 8; ++q) {
      v2f a0, a1, b0, b1;
      a0.x = a0p[q * 4];  a0.y = a0p[q * 4 + 1];
      a1.x = a1p[q * 4];  a1.y = a1p[q * 4 + 1];
      b0.x = b0p[q * 4];  b0.y = b0p[q * 4 + 1];
      b1.x = b1p[q * 4];  b1.y = b1p[q * 4 + 1];
      c00 = __builtin_amdgcn_wmma_f32_16x16x4_f32(false, a0, false, b0, (short)0, c00, false, false);
      c01 = __builtin_amdgcn_wmma_f32_16x16x4_f32(false, a0, false, b1, (short)0, c01, false, false);
      c10 = __builtin_amdgcn_wmma_f32_16x16x4_f32(false, a1, false, b0, (short)0, c10, false, false);
      c11 = __builtin_amdgcn_wmma_f32_16x16x4_f32(false, a1, false, b1, (short)0, c11, false, false);
    }
  }

  // ---- s_min: cost = (Sa - 2C + Sb) / 1024, min over valid (dy,dx) ----
  float lmin = FLT_MAX;
  const float inv = 1.0f / 1024.0f;
#pragma unroll
  for (int f = 0; f < 4; ++f) {
    v8f cf = (f == 0) ? c00 : (f == 1) ? c01 : (f == 2) ? c10 : c11;
    int mi = f >> 1, ni = f & 1;
#pragma unroll
    for (int j = 0; j < 8; ++j) {
      int m = mi * 16 + j + half * 8;   // dy
      int n = ni * 16 + lid;            // dx
      if (m < NSH && n < NSH) {
        int rs = min(max(ty * 32 + m - 8, 0), 96) - rbase0;
        int cs = min(max(tx * 32 + n - 8, 0), 96) - cbase0;
        float sb = sat[(rs + 32) * SATP + (cs + 32)] - sat[rs * SATP + (cs + 32)]
                 - sat[(rs + 32) * SATP + cs]        + sat[rs * SATP + cs];
        float cost = (sa - 2.0f * cf[j] + sb) * inv;
        lmin = fminf(lmin, cost);
      }
    }
  }
  float smin = wave_min32(lmin);

  // ---- rotation branch: 11 angles, bilinear with validity mask ----
  float rmin = FLT_MAX;
  for (int a = 0; a < NANG; ++a) {
    double rad = (double)(a - 5) * 0.017453292519943295;
    double al = cos(rad), be = sin(rad);
    float num = 0.f, den = 0.f;
    for (int it = 0; it < 32; ++it) {
      int p  = it * 32 + lane;
      int py = p >> 5, px = p & 31;
      double xd = (double)px - 16.0, yd = (double)py - 16.0;
      double sx = al * xd - be * yd + 16.0;
      double sy = be * xd + al * yd + 16.0;
      double fx0 = floor(sx), fy0 = floor(sy);
      int   x0 = (int)fx0, y0 = (int)fy0;
      float fx = (float)(sx - fx0), fy = (float)(sy - fy0);
      float rot = 0.f, msk = 0.f;
#pragma unroll
      for (int t = 0; t < 4; ++t) {
        int dx = t & 1, dy = t >> 1;
        int xi = x0 + dx, yi = y0 + dy;
        float wgt = (dx ? fx : 1.f - fx) * (dy ? fy : 1.f - fy);
        bool valid = (xi >= 0) & (xi < 32) & (yi >= 0) & (yi < 32);
        float w = valid ? wgt : 0.f;
        int xc = min(max(xi, 0), 31), yc = min(max(yi, 0), 31);
        rot += w * slab[(rc0 + yc) * SLABP + (cc0 + xc)];
        msk += w;
      }
      float d = aPad[(py + 16) * APADP + px] - rot;
      num += d * d * msk;
      den += msk;
    }
    num = wave_sum32(num);
    den = wave_sum32(den);
    rmin = fminf(rmin, num / den);
  }

  if (lane == 0) subMin[bt] = fminf(smin, rmin);
}

// Per batch: sum of the 8 smallest of its 16 tile minima.
__global__ __launch_bounds__(32)
void topk_kernel(const float* __restrict__ subMin, float* __restrict__ out)
{
  int b = threadIdx.x;
  if (b >= 16) return;
  float v[16];
#pragma unroll
  for (int t = 0; t < 16; ++t) v[t] = subMin[b * 16 + t];
  float s = 0.f;
#pragma unroll
  for (int k = 0; k < 8; ++k) {
    int jm = 0; float mv = v[0];
#pragma unroll
    for (int t = 1; t < 16; ++t) if (v[t] < mv) { mv = v[t]; jm = t; }
    s += mv;
    v[jm] = FLT_MAX;
  }
  out[b] = s;
}

extern "C" void kernel_launch(void* const* d_in, const int* in_sizes, int n_in,
                              void* d_out, int out_size, void* d_ws, size_t ws_size,
                              hipStream_t stream) {
  (void)in_sizes; (void)n_in; (void)out_size; (void)ws_size;
  const float* fm1 = (const float*)d_in[0];
  const float* fm2 = (const float*)d_in[1];
  float* out    = (float*)d_out;
  float* subMin = (float*)d_ws;          // 256 floats of scratch

  rip_tile_kernel<<<256, 32, 0, stream>>>(fm1, fm2, subMin);
  topk_kernel<<<1, 32, 0, stream>>>(subMin, out);
}